// Biattention_14448269984082
// MI455X (gfx1250) — compile-verified
//
#include <hip/hip_runtime.h>

typedef float v2f __attribute__((ext_vector_type(2)));
typedef float v8f __attribute__((ext_vector_type(8)));
typedef int   v4i __attribute__((ext_vector_type(4)));

// Problem shape (fixed by the reference)
constexpr int S_LEN = 2048;
constexpr int D_DIM = 1024;
constexpr int OUT_D = 5 * D_DIM;
constexpr int QT    = 16;              // query-tile rows per workgroup
constexpr int QS_STR = D_DIM + 4;      // 1028: bank stride 4 -> conflict-free A reads
constexpr int PS_STR = S_LEN + 4;      // 2052: bank stride 4 -> conflict-free A reads

__global__ __launch_bounds__(256, 1)
void biattn_fused_kernel(const float* __restrict__ x,
                         const int*   __restrict__ mask,
                         float*       __restrict__ out) {
    // LDS: Q tile (16 x 1024) + full score/prob row block (16 x 2048)
    __shared__ float Qs[QT * QS_STR];      //  65,792 B
    __shared__ float Ps[QT * PS_STR];      // 131,328 B  (total ~197 KB < 320 KB/WGP)

    const int wg   = blockIdx.x;
    const int b    = wg >> 7;                  // / (S/16) = /128
    const int qt   = wg & 127;
    const int q0   = qt * QT;
    const int tid  = threadIdx.x;
    const int lane = tid & 31;
    const int wave = tid >> 5;

    const float* xb = x + (size_t)b * S_LEN * D_DIM;

    // ---------------- Stage Q tile into LDS ----------------
    // CDNA5 async global->LDS path (ASYNCcnt): global_load_async_to_lds_b128.
#if __has_builtin(__builtin_amdgcn_global_load_async_to_lds_b128)
    for (int i = tid; i < QT * (D_DIM / 4); i += 256) {
        const int r = i >> 8;                  // D_DIM/4 == 256 vec4 per row
        const int c = (i & 255) * 4;
        const float* g = xb + (size_t)(q0 + r) * D_DIM + c;
        float* l = &Qs[r * QS_STR + c];
        __builtin_amdgcn_global_load_async_to_lds_b128(
            (__attribute__((address_space(1))) v4i*)g,
            (__attribute__((address_space(3))) v4i*)l, 0, 0);
    }
#if __has_builtin(__builtin_amdgcn_s_wait_asynccnt)
    __builtin_amdgcn_s_wait_asynccnt(0);
#else
    asm volatile("s_wait_asynccnt 0" ::: "memory");
#endif
#else
    for (int i = tid; i < QT * (D_DIM / 4); i += 256) {
        const int r = i >> 8;
        const int c = (i & 255) * 4;
        const float4 v = *(const float4*)(xb + (size_t)(q0 + r) * D_DIM + c);
        float* dst = &Qs[r * QS_STR + c];
        dst[0] = v.x; dst[1] = v.y; dst[2] = v.z; dst[3] = v.w;
    }
#endif
    __syncthreads();

    // WMMA f32 16x16x4 operand indexing (wave32):
    //   A (16x4): M = lane&15 for all lanes; VGPR0/1 hold K = koff, koff+1
    //             where koff = (lane<16) ? 0 : 2
    //   B (4x16): N = lane&15; VGPR0/1 hold K = koff, koff+1
    //   C/D (16x16): N = lane&15; VGPR r holds M = r + (lane<16 ? 0 : 8)
    const int n     = lane & 15;
    const int koff  = (lane < 16) ? 0 : 2;
    const int qbase = (lane < 16) ? 0 : 8;

    // ------- Score pass: S = Q . K^T, 2 key-tiles per wave iteration --------
    // One A (Q) fragment feeds two WMMAs -> halves LDS A traffic.
    for (int p = 0; p < 8; ++p) {
        const int kt0  = p * 16 + wave * 2;    // two adjacent key tiles
        const int key0 = kt0 * 16;             // second tile at key0 + 16
        const float* aptr = &Qs[n * QS_STR + koff];
        const float* bptr = xb + (size_t)(key0 + n) * D_DIM + koff;

        v8f c0 = {}, c1 = {};
#pragma unroll 4
        for (int k0 = 0; k0 < D_DIM; k0 += 4) {
            v2f a   = *(const v2f*)(aptr + k0);                        // ds_load_b64
            v2f bm0 = *(const v2f*)(bptr + k0);                        // key tile 0
            v2f bm1 = *(const v2f*)(bptr + k0 + 16 * (size_t)D_DIM);   // key tile 1 (+64KB imm)
            c0 = __builtin_amdgcn_wmma_f32_16x16x4_f32(
                     false, a, false, bm0, (short)0, c0, false, false);
            c1 = __builtin_amdgcn_wmma_f32_16x16x4_f32(
                     false, a, false, bm1, (short)0, c1, false, false);
        }

        const int m0 = mask[b * S_LEN + key0 + n];
        const int m1 = mask[b * S_LEN + key0 + 16 + n];
#pragma unroll
        for (int r = 0; r < 8; ++r) {
            const float v0 = (m0 == 0) ? -__builtin_inff() : c0[r];
            const float v1 = (m1 == 0) ? -__builtin_inff() : c1[r];
            Ps[(qbase + r) * PS_STR + key0 + n]      = v0;
            Ps[(qbase + r) * PS_STR + key0 + 16 + n] = v1;
        }
    }
    __syncthreads();

    // ---------------- Exact softmax over keys (2 rows per wave) ------------
    for (int q = wave; q < QT; q += 8) {
        float* row = &Ps[q * PS_STR];
        float mx = -__builtin_inff();
        for (int k = lane; k < S_LEN; k += 32) mx = fmaxf(mx, row[k]);
#pragma unroll
        for (int off = 16; off > 0; off >>= 1) mx = fmaxf(mx, __shfl_xor(mx, off, 32));
        float sum = 0.0f;
        for (int k = lane; k < S_LEN; k += 32) {
            const float e = __expf(row[k] - mx);
            row[k] = e;
            sum += e;
        }
#pragma unroll
        for (int off = 16; off > 0; off >>= 1) sum += __shfl_xor(sum, off, 32);
        const float inv = 1.0f / sum;
        for (int k = lane; k < S_LEN; k += 32) row[k] *= inv;
    }
    __syncthreads();

    // ------- Context pass: C = P . V, 2 D-tiles per iteration, NT stores ----
    const int d0base = wave * 128;                 // each wave owns 128 D columns
    const size_t obase = ((size_t)b * S_LEN + q0) * OUT_D;

    for (int t = 0; t < 4; ++t) {
        const int d0 = d0base + t * 32;            // two tiles: d0, d0+16
        const float* aptr = &Ps[n * PS_STR + koff];
        const float* bcol = xb + (size_t)koff * D_DIM + d0 + n;

        v8f acc0 = {}, acc1 = {};
#pragma unroll 4
        for (int k0 = 0; k0 < S_LEN; k0 += 4) {
            v2f a = *(const v2f*)(aptr + k0);      // ds_load_b64 (shared by 2 WMMAs)
            v2f bm0, bm1;
            bm0.x = bcol[(size_t)k0 * D_DIM];           // V[k0+koff  ][d0+n]
            bm0.y = bcol[(size_t)(k0 + 1) * D_DIM];     // V[k0+koff+1][d0+n]
            bm1.x = bcol[(size_t)k0 * D_DIM + 16];      // V[k0+koff  ][d0+16+n]
            bm1.y = bcol[(size_t)(k0 + 1) * D_DIM + 16];
            acc0 = __builtin_amdgcn_wmma_f32_16x16x4_f32(
                       false, a, false, bm0, (short)0, acc0, false, false);
            acc1 = __builtin_amdgcn_wmma_f32_16x16x4_f32(
                       false, a, false, bm1, (short)0, acc1, false, false);
        }

        // Fused epilogue: stream all 5 output chunks with non-temporal stores
        // (keeps x L2-resident; output is write-once).
#pragma unroll
        for (int r = 0; r < 8; ++r) {
            const int q = qbase + r;
            float* o = out + obase + (size_t)q * OUT_D + d0 + n;
            {
                const float cv = acc0[r];
                const float xv = Qs[q * QS_STR + d0 + n];
                __builtin_nontemporal_store(xv,        o);
                __builtin_nontemporal_store(cv,        o + D_DIM);
                __builtin_nontemporal_store(xv + cv,   o + 2 * D_DIM);
                __builtin_nontemporal_store(xv - cv,   o + 3 * D_DIM);
                __builtin_nontemporal_store(xv * cv,   o + 4 * D_DIM);
            }
            {
                const float cv = acc1[r];
                const float xv = Qs[q * QS_STR + d0 + 16 + n];
                float* o1 = o + 16;
                __builtin_nontemporal_store(xv,        o1);
                __builtin_nontemporal_store(cv,        o1 + D_DIM);
                __builtin_nontemporal_store(xv + cv,   o1 + 2 * D_DIM);
                __builtin_nontemporal_store(xv - cv,   o1 + 3 * D_DIM);
                __builtin_nontemporal_store(xv * cv,   o1 + 4 * D_DIM);
            }
        }
    }
}

extern "C" void kernel_launch(void* const* d_in, const int* in_sizes, int n_in,
                              void* d_out, int out_size, void* d_ws, size_t ws_size,
                              hipStream_t stream) {
    const float* x    = (const float*)d_in[0];
    const int*   mask = (const int*)d_in[1];
    float*       out  = (float*)d_out;

    const int B = in_sizes[0] / (S_LEN * D_DIM);   // = 8
    dim3 grid(B * (S_LEN / QT));                   // 8 * 128 = 1024 workgroups
    dim3 block(256);                               // 8 waves (wave32)
    biattn_fused_kernel<<<grid, block, 0, stream>>>(x, mask, out);
}